// ParametricInterpolation_68453188764168
// MI455X (gfx1250) — compile-verified
//
#include <hip/hip_runtime.h>
#include <hip/hip_bf16.h>
#include <math.h>

#define SIG 2048
#define BATCH 16384

typedef __attribute__((ext_vector_type(2))) float v2f;
typedef __attribute__((ext_vector_type(8))) float v8f;

// ---------------- workspace layout ----------------
// [0, 16MB)          : Kmat  f32[2048*2048]  (materialized sparse kernel)
// [+0]               : bitmap u32[512]       (128 rb x 128 cb nonzero-tile bits)
// [+2048B]           : counts i32[128]       (row-blocks per column-block)
// [+2560B]           : lists  i32[128*128]   (row-block indices per column-block)
#define KMAT_WORDS (SIG * SIG)
#define BITMAP_WORDS 512

// Zero Kmat + bitmap (contiguous) every call: deterministic rebuild.
__global__ void pi_zero(unsigned int* __restrict__ w, int n) {
    int i = blockIdx.x * blockDim.x + threadIdx.x;
    int stride = gridDim.x * blockDim.x;
    for (; i < n; i += stride) w[i] = 0u;
}

// One thread per signal index: evaluate the quartic exactly like the
// reference (f32 ops, rintf == round-half-to-even), scatter the two
// weights into row i of Kmat (row-private -> no races), and mark the
// touched 16x16 tiles in the bitmap (atomicOr: order-independent).
__global__ void pi_fill(const float* __restrict__ params,
                        float* __restrict__ Kmat,
                        unsigned int* __restrict__ bitmap) {
    int i = blockIdx.x * blockDim.x + threadIdx.x;
    if (i >= SIG) return;
    float p0 = params[0] * 1e13f;
    float p1 = params[1] * 1e9f;
    float p2 = params[2] * 1e5f;
    float p3 = params[3] * 10.0f;
    float p4 = params[4] * 100.0f;
    float t = (float)i;
    float t2 = t * t;
    float t3 = t2 * t;
    float t4 = t2 * t2;
    float curve = p0 * t4 + p1 * t3 + p2 * t2 + p3 * t + p4;
    float ci = rintf(curve);                  // round half-to-even, matches jnp.round
    float k = ci - curve;                     // signed fractional offset
    bool valid = (ci >= -2.0e9f) && (ci <= 2.0e9f);   // also false for NaN
    int pos = valid ? (int)ci : -1000000;     // sentinel -> both scatters dropped
    int rb = i >> 4;
    int c0 = pos - 1;
    if (c0 >= 0 && c0 < SIG) {
        Kmat[(size_t)i * SIG + c0] = k;
        int cb = c0 >> 4;
        atomicOr(&bitmap[rb * 4 + (cb >> 5)], 1u << (cb & 31));
    }
    if (pos >= 0 && pos < SIG) {
        Kmat[(size_t)i * SIG + pos] = 1.0f - k;
        int cb = pos >> 4;
        atomicOr(&bitmap[rb * 4 + (cb >> 5)], 1u << (cb & 31));
    }
}

// One thread per column-block: collect contributing row-blocks in
// ascending order -> deterministic accumulation order in the matmul.
__global__ void pi_lists(const unsigned int* __restrict__ bitmap,
                         int* __restrict__ counts,
                         int* __restrict__ lists) {
    int cb = blockIdx.x * blockDim.x + threadIdx.x;
    if (cb >= SIG / 16) return;
    int n = 0;
    unsigned int mask = 1u << (cb & 31);
    int word = cb >> 5;
    for (int rb = 0; rb < SIG / 16; ++rb) {
        if (bitmap[rb * 4 + word] & mask) lists[cb * 128 + n++] = rb;
    }
    counts[cb] = n;
}

// Block-sparse banded GEMM: one wave32 per 16x16 output tile.
// Chains 4x V_WMMA_F32_16X16X4_F32 per contributing 16x16 K-tile.
__global__ __launch_bounds__(256) void pi_wmma(
        const float* __restrict__ x,
        const float* __restrict__ Kmat,
        const int* __restrict__ counts,
        const int* __restrict__ lists,
        float* __restrict__ out) {
    const int lane = threadIdx.x;          // 0..31 (wave32)
    const int wave = threadIdx.y;          // 0..7
    const int cb = blockIdx.x;             // column block 0..127
    const int bb = blockIdx.y * 8 + wave;  // batch-row block 0..1023
    const int half = lane >> 4;            // 0: lanes 0-15, 1: lanes 16-31
    const int m = lane & 15;
    const int c0 = cb * 16;
    const int b0 = bb * 16;

    v8f acc = {};                          // 16x16 f32 C/D tile (8 VGPRs)
    const int cnt = counts[cb];
    const int* __restrict__ lst = lists + cb * 128;
    const float* __restrict__ xrow = x + (size_t)(b0 + m) * SIG;

    for (int e = 0; e < cnt; ++e) {
        const int i0 = lst[e] * 16;
        if (e + 1 < cnt) {
            // hint the next row-block's A tile (global_prefetch_b8)
            __builtin_prefetch(xrow + lst[e + 1] * 16, 0, 1);
        }
#pragma unroll
        for (int j = 0; j < 4; ++j) {
            const int kk = i0 + 4 * j + 2 * half;
            // A 16x4: lane m holds row M=m, K = kk (VGPR0) and kk+1 (VGPR1)
            v2f a = *(const v2f*)(xrow + kk);
            // B 4x16: lane m holds col N=m, rows K = kk (VGPR0) and kk+1 (VGPR1)
            v2f b;
            b.x = Kmat[(size_t)kk * SIG + c0 + m];
            b.y = Kmat[(size_t)(kk + 1) * SIG + c0 + m];
            acc = __builtin_amdgcn_wmma_f32_16x16x4_f32(
                /*neg_a=*/false, a, /*neg_b=*/false, b,
                /*c_mod=*/(short)0, acc, /*reuse_a=*/false, /*reuse_b=*/false);
        }
    }

    // C/D layout: VGPR r -> row M = r + 8*half, col N = m
#pragma unroll
    for (int r = 0; r < 8; ++r) {
        out[(size_t)(b0 + r + 8 * half) * SIG + c0 + m] = acc[r];
    }
}

extern "C" void kernel_launch(void* const* d_in, const int* in_sizes, int n_in,
                              void* d_out, int out_size, void* d_ws, size_t ws_size,
                              hipStream_t stream) {
    const float* x = (const float*)d_in[0];        // (16384, 2048) f32
    const float* params = (const float*)d_in[1];   // (5,) f32
    float* out = (float*)d_out;                    // (16384, 2048) f32

    char* ws = (char*)d_ws;
    float* Kmat = (float*)ws;
    unsigned int* bitmap = (unsigned int*)(ws + (size_t)KMAT_WORDS * 4);
    int* counts = (int*)(ws + (size_t)KMAT_WORDS * 4 + BITMAP_WORDS * 4);
    int* lists = (int*)(ws + (size_t)KMAT_WORDS * 4 + BITMAP_WORDS * 4 + 128 * 4);

    // Phase 1a: zero Kmat + bitmap (contiguous words)
    {
        int n = KMAT_WORDS + BITMAP_WORDS;
        int threads = 256;
        int blocks = 2048;
        pi_zero<<<blocks, threads, 0, stream>>>((unsigned int*)ws, n);
    }
    // Phase 1b: evaluate curve, scatter weights, mark tiles
    pi_fill<<<(SIG + 255) / 256, 256, 0, stream>>>(params, Kmat, bitmap);
    // Phase 1c: per-column-block row-block lists (ascending -> deterministic)
    pi_lists<<<1, 128, 0, stream>>>(bitmap, counts, lists);

    // Phase 2: block-sparse WMMA matmul, one wave per 16x16 out tile
    dim3 block(32, 8);
    dim3 grid(SIG / 16, (BATCH / 16) / 8);
    pi_wmma<<<grid, block, 0, stream>>>(x, Kmat, counts, lists, out);
}